// gru_module_4_10264971837858
// MI455X (gfx1250) — compile-verified
//
#include <hip/hip_runtime.h>

// ---------------------------------------------------------------------------
// Fused GRU-like module for MI455X (gfx1250, wave32).
// One wave per batch element; WMMA f32 16x16x4 for the two 24x24 matvecs;
// LDS staging of the per-b matrices with async-to-LDS coalesced loads.
// ---------------------------------------------------------------------------

typedef __attribute__((ext_vector_type(2))) float v2f;
typedef __attribute__((ext_vector_type(8))) float v8f;
typedef __attribute__((ext_vector_type(4))) int   v4i;

#ifndef __has_builtin
#define __has_builtin(x) 0
#endif

#if __has_builtin(__builtin_amdgcn_wmma_f32_16x16x4_f32)
#define HAVE_WMMA_F32_K4 1
#endif

#if __has_builtin(__builtin_amdgcn_global_load_async_to_lds_b128)
#define HAVE_ASYNC_LDS 1
#endif

#define ETA_C 0.95f
#define ETA_D 0.9f
#define T_DIM 24
#define WPB 8                 // waves (batch elements) per block
#define MAT_PAD 592           // 576 matrix floats + 16 zeroed pad (B-tile overread)
#define LDS_PER_WAVE (2 * MAT_PAD)

// Computes d[s] = sum_t m[t*24+s] * dd(t) for a 24x24 row-major matrix in LDS.
// Returns: lo = d[s] for s = lane (valid lanes 0..15)
//          hi = d[16+n] on lane n (valid lanes 0..7)
__device__ __forceinline__ void matvec24(const float* m, float ddv, int lane,
                                         float& lo, float& hi) {
#if defined(HAVE_WMMA_F32_K4)
    const int srcAdd = (lane >> 4) << 1;   // K split: lanes 0-15 -> +0, 16-31 -> +2
    const int n = lane & 15;
    v8f accA = {0.f, 0.f, 0.f, 0.f, 0.f, 0.f, 0.f, 0.f};
    v8f accB = {0.f, 0.f, 0.f, 0.f, 0.f, 0.f, 0.f, 0.f};
#pragma unroll
    for (int t0 = 0; t0 < 24; t0 += 4) {
        // A (16x4 f32): broadcast dd across all 16 M rows.
        // Layout: VGPR0 = K0 (lanes 0-15) / K2 (lanes 16-31); VGPR1 = K1 / K3.
        v2f Aop;
        Aop.x = __shfl(ddv, t0 + srcAdd, 32);
        Aop.y = __shfl(ddv, t0 + srcAdd + 1, 32);
        // B (4x16 f32): VGPR0 = row K0 (lanes 0-15) / K2 (lanes 16-31); VGPR1 = K1/K3.
        const int r0 = (t0 + srcAdd) * 24 + n;
        const int r1 = r0 + 24;
        v2f B0, B1;
        B0.x = m[r0];       B0.y = m[r1];        // column tile s0 = 0
        B1.x = m[r0 + 16];  B1.y = m[r1 + 16];   // column tile s0 = 16
        accA = __builtin_amdgcn_wmma_f32_16x16x4_f32(
            false, Aop, false, B0, (short)0, accA, false, false);
        accB = __builtin_amdgcn_wmma_f32_16x16x4_f32(
            false, Aop, false, B1, (short)0, accB, false, false);
    }
    lo = accA[0];   // D row 0: VGPR0, lanes 0-15 = N
    hi = accB[0];
#else
    // VALU fallback (kept so the TU always compiles; histogram tells us which
    // path was taken).
    const int s = (lane < T_DIM) ? lane : 0;
    float acc = 0.f;
#pragma unroll
    for (int t = 0; t < 24; ++t) {
        float v = __shfl(ddv, t, 32);
        acc = fmaf(m[t * 24 + s], v, acc);
    }
    lo = acc;                                   // lanes 0..15: s = lane
    hi = __shfl(acc, 16 + (lane & 15), 32);     // lane n: s = 16+n
#endif
}

__global__ __launch_bounds__(WPB * 32) void gru_fused_kernel(
    const float* __restrict__ X,      // (B,24)
    const float* __restrict__ Xm1,    // (B,24,24)
    const float* __restrict__ Xm2,    // (B,3,24,3)
    const float* __restrict__ Xm3,    // (B,24,24)
    const float* __restrict__ C,      // (B,24,3)
    float* __restrict__ Y,            // (B,5,24)
    int Btot)
{
    __shared__ float lds[WPB * LDS_PER_WAVE];
    const int lane = (int)(threadIdx.x & 31);
    const int wave = (int)(threadIdx.x >> 5);
    const int b = (int)blockIdx.x * WPB + wave;
    if (b >= Btot) return;   // whole wave uniform exit

    float* m1 = &lds[wave * LDS_PER_WAVE];
    float* m3 = m1 + MAT_PAD;
    const float* g1 = Xm1 + (size_t)b * 576;
    const float* g3 = Xm3 + (size_t)b * 576;

    // ---- stage both 24x24 matrices into LDS, fully coalesced 128b ----
#if defined(HAVE_ASYNC_LDS)
    typedef __attribute__((address_space(3))) v4i lds_v4i;
#pragma unroll
    for (int k = 0; k < 4; ++k) {
        const int i = lane + 32 * k;
        __builtin_amdgcn_global_load_async_to_lds_b128(
            (v4i*)(g1 + i * 4), (lds_v4i*)(m1 + i * 4), 0, 0);
        __builtin_amdgcn_global_load_async_to_lds_b128(
            (v4i*)(g3 + i * 4), (lds_v4i*)(m3 + i * 4), 0, 0);
    }
    if (lane < 16) {
        const int i = lane + 128;
        __builtin_amdgcn_global_load_async_to_lds_b128(
            (v4i*)(g1 + i * 4), (lds_v4i*)(m1 + i * 4), 0, 0);
        __builtin_amdgcn_global_load_async_to_lds_b128(
            (v4i*)(g3 + i * 4), (lds_v4i*)(m3 + i * 4), 0, 0);
    }
#else
#pragma unroll
    for (int k = 0; k < 4; ++k) {
        const int i = lane + 32 * k;
        float4 v1 = *(const float4*)(g1 + i * 4);
        float4 v3 = *(const float4*)(g3 + i * 4);
        *(float4*)(m1 + i * 4) = v1;
        *(float4*)(m3 + i * 4) = v3;
    }
    if (lane < 16) {
        const int i = lane + 128;
        float4 v1 = *(const float4*)(g1 + i * 4);
        float4 v3 = *(const float4*)(g3 + i * 4);
        *(float4*)(m1 + i * 4) = v1;
        *(float4*)(m3 + i * 4) = v3;
    }
#endif
    if (lane < 16) { m1[576 + lane] = 0.f; m3[576 + lane] = 0.f; }

    // ---- phase 1: per-t elementwise chain (lane t; lanes 24-31 duplicate) ----
    const int t = (lane < T_DIM) ? lane : (T_DIM - 1);
    const float* Cb = C   + (size_t)b * 72  + t * 3;
    const float* A0 = Xm2 + (size_t)b * 216 + t * 3;
    const float c0 = Cb[0], c1 = Cb[1], c2 = Cb[2];
    const float a00 = A0[0],   a01 = A0[1],   a02 = A0[2];
    const float a10 = A0[72],  a11 = A0[73],  a12 = A0[74];
    const float a20 = A0[144], a21 = A0[145], a22 = A0[146];
    const float x = X[(size_t)b * 24 + t];

    const float delta0 = a00 * c0 + a01 * c1 + a02 * c2;
    const float d0 = delta0 - fmaxf(delta0 - x, 0.f);
    const float r1v = x - d0;
    const float delta1 = a10 * c0 + a11 * c1 + a12 * c2;
    const float d1 = delta1 - fmaxf(delta1 - r1v, 0.f);
    const float r2v = r1v - d1;
    const float delta2 = a20 * c0 + a21 * c1 + a22 * c2;
    const float d2 = delta2 - fmaxf(delta2 - r2v, 0.f);

    const float dg = a02 * d0 + a12 * d1 + a22 * d2;
    const float dp = a01 * d0 + a11 * d1 + a21 * d2;
    const float dd = (a00 * d0 + a10 * d1 + a20 * d2) / ETA_D;

    float* Yb = Y + (size_t)b * 120;
    if (lane < T_DIM) {
        Yb[24 + lane] = dp;   // Y[:,1] = d_p
        Yb[96 + lane] = dd;   // Y[:,4] = d_d
    }

    // ---- make staged LDS data visible / ordered ----
#if defined(HAVE_ASYNC_LDS)
    asm volatile("s_wait_asynccnt 0" ::: "memory");
#endif
    __builtin_amdgcn_wave_barrier();

    // ---- phase 2: d_c = (dd @ Xm1)/ETA_C ; d_s = dd @ Xm3 via WMMA ----
    float loC, hiC, loS, hiS;
    matvec24(m1, dd, lane, loC, hiC);
    matvec24(m3, dd, lane, loS, hiS);

    const float dcA = loC / ETA_C;
    const float dcB = hiC / ETA_C;
    // gather d_c[lane] for lanes 16..23 from dcB lanes 0..7
    const float dcHi = __shfl(dcB, (lane - 16) & 31, 32);
    const float dcv = (lane < 16) ? dcA : dcHi;

    if (lane < T_DIM) Yb[lane] = dg + dcv;   // Y[:,0] = d_g + d_c
    if (lane < 16)    Yb[72 + lane] = dcA;   // Y[:,3] = d_c
    if (lane < 8)     Yb[88 + lane] = dcB;
    if (lane < 16)    Yb[48 + lane] = loS;   // Y[:,2] = d_s
    if (lane < 8)     Yb[64 + lane] = hiS;
}

extern "C" void kernel_launch(void* const* d_in, const int* in_sizes, int n_in,
                              void* d_out, int out_size, void* d_ws, size_t ws_size,
                              hipStream_t stream) {
    (void)n_in; (void)out_size; (void)d_ws; (void)ws_size;
    const float* X   = (const float*)d_in[0];
    const float* Xm1 = (const float*)d_in[1];
    const float* Xm2 = (const float*)d_in[2];
    const float* Xm3 = (const float*)d_in[3];
    const float* C   = (const float*)d_in[4];
    float* Y = (float*)d_out;

    const int Btot = in_sizes[0] / T_DIM;
    const int blocks = (Btot + WPB - 1) / WPB;
    gru_fused_kernel<<<blocks, WPB * 32, 0, stream>>>(X, Xm1, Xm2, Xm3, C, Y, Btot);
}